// CTCDecodeLayer_8186207666834
// MI455X (gfx1250) — compile-verified
//
#include <hip/hip_runtime.h>
#include <stdint.h>

// Problem constants (match reference)
#define BB    32
#define TT    1024
#define VV    6000
#define BLANK (VV - 1)

// Tiling for kernel 1
#define ROWS  8      // (b,t) rows per block, one wave each
#define CHUNK 1000   // floats per row per TDM tile (6 * 1000 == 6000)
#define NC    6

typedef __attribute__((ext_vector_type(4))) unsigned int v4u;
typedef __attribute__((ext_vector_type(8))) int          v8i;
typedef __attribute__((ext_vector_type(4))) int          v4i;

#if __has_include(<hip/amd_detail/amd_gfx1250_TDM.h>)
#define TDM_6ARG 1
#endif

// Issue one TDM 2D tile load: 8 rows x CHUNK floats, row stride VV floats,
// contiguous into LDS at lds_off. Descriptor per CDNA5 ISA.
__device__ __forceinline__ void tdm_load_tile(uint32_t lds_off, uint64_t gaddr) {
  v4u g0;
  g0.x = 1u;                                               // count=1 (valid), no gather
  g0.y = lds_off;                                          // lds_addr (bytes)
  g0.z = (uint32_t)gaddr;                                  // global_addr[31:0]
  g0.w = (uint32_t)((gaddr >> 32) & 0x01FFFFFFu) | (2u << 30); // addr[56:32] | type=2

  v8i g1;
  g1[0] = 0x00020000;                // data_size=2 (4 bytes); mask=0; no pad/iter/atomic
  g1[1] = 0;                         // atomic_barrier_addr=0; tensor_dim0[15:0]=0
  g1[2] = 0x00004000;                // tensor_dim0[31:16]=0x4000 (dim0=1<<30, no OOB clip)
  g1[3] = (CHUNK << 16) | 0x4000;    // tile_dim0=1000; tensor_dim1[31:16]=0x4000 (1<<30)
  g1[4] = ROWS;                      // tile_dim1=8; tile_dim2=0
  g1[5] = VV;                        // tensor_dim0_stride = 6000 elements
  g1[6] = 0;                         // stride0 hi / stride1 lo
  g1[7] = 0;                         // stride1 hi

  v4i z4 = {0, 0, 0, 0};             // groups 2/3 unused (2D tensor)
#ifdef TDM_6ARG
  v8i z8 = {0, 0, 0, 0, 0, 0, 0, 0};
  __builtin_amdgcn_tensor_load_to_lds(g0, g1, z4, z4, z8, 0);
#else
  __builtin_amdgcn_tensor_load_to_lds(g0, g1, z4, z4, 0);
#endif
}

// Kernel 1: per-(b,t) argmax + maxlog over V, TDM double-buffered through LDS.
// grid = (BB*TT)/ROWS blocks of 256 threads (8 waves; wave w owns row r0+w).
__global__ __launch_bounds__(256) void ctc_rowmax_kernel(
    const float* __restrict__ pred, int* __restrict__ bestOut,
    float* __restrict__ mlOut) {
  __shared__ float smem[2][ROWS][CHUNK];   // 64,000 bytes, double buffered

  const int tid  = threadIdx.x;
  const int w    = tid >> 5;
  const int lane = tid & 31;
  const long long r0 = (long long)blockIdx.x * ROWS;
  const uint64_t gbase =
      (uint64_t)(uintptr_t)pred + (uint64_t)r0 * (uint64_t)VV * 4ull;

  // Scalar (SCC) branch so waves 1..7 skip the TDM issue entirely
  // (tensor ops ignore EXEC, so an exec-masked path could double-issue).
  const int wave_scalar = __builtin_amdgcn_readfirstlane(tid) >> 5;

  if (wave_scalar == 0) {
    tdm_load_tile((uint32_t)(uintptr_t)&smem[0][0][0], gbase);
  }

  float maxp = -1.0f;   // probabilities are in [0,1]
  int   maxi = 0;

  for (int c = 0; c < NC; ++c) {
    if (wave_scalar == 0) {
      if (c + 1 < NC) {
        tdm_load_tile((uint32_t)(uintptr_t)&smem[(c + 1) & 1][0][0],
                      gbase + (uint64_t)(c + 1) * CHUNK * 4ull);
        __builtin_amdgcn_s_wait_tensorcnt((short)1);  // chunk c landed, c+1 in flight
      } else {
        __builtin_amdgcn_s_wait_tensorcnt((short)0);
      }
    }
    __syncthreads();  // chunk c visible in LDS to all waves

    const float4* rowp =
        reinterpret_cast<const float4*>(&smem[c & 1][w][0]);  // 250 float4s
    for (int q = lane; q < CHUNK / 4; q += 32) {
      float4 v = rowp[q];
      int base = c * CHUNK + q * 4;
      if (v.x > maxp) { maxp = v.x; maxi = base; }      // strict > => lowest index
      if (v.y > maxp) { maxp = v.y; maxi = base + 1; }  // wins ties within a lane
      if (v.z > maxp) { maxp = v.z; maxi = base + 2; }
      if (v.w > maxp) { maxp = v.w; maxi = base + 3; }
    }
    __syncthreads();  // everyone done reading buf[c&1] before it is refilled
  }

  // wave32 butterfly reduction: max value, lowest index on ties
  for (int off = 16; off > 0; off >>= 1) {
    float op = __shfl_xor(maxp, off, 32);
    int   oi = __shfl_xor(maxi, off, 32);
    if (op > maxp || (op == maxp && oi < maxi)) { maxp = op; maxi = oi; }
  }

  if (lane == 0) {
    long long row = r0 + w;
    bestOut[row] = maxi;
    mlOut[row]   = logf(maxp + 1e-7f);
  }
}

// Kernel 2: per-batch blank/repeat removal + compaction + logprob sum.
// grid = BB blocks of one wave32.
__global__ __launch_bounds__(32) void ctc_decode_kernel(
    const int* __restrict__ best, const float* __restrict__ ml,
    const int* __restrict__ in_len, float* __restrict__ out) {
  const int b    = blockIdx.x;
  const int lane = threadIdx.x;
  const int len  = in_len[b];            // in_len is (B,1)
  float* dec = out + (size_t)b * TT;

  for (int t = lane; t < TT; t += 32) dec[t] = -1.0f;  // default fill

  int   prev  = -1;
  int   count = 0;
  float suml  = 0.0f;
  const int rowbase = b * TT;

  for (int t0 = 0; t0 < TT; t0 += 32) {
    int   t  = t0 + lane;
    int   bi = best[rowbase + t];
    float m  = ml[rowbase + t];
    bool valid = t < len;

    int pv = __shfl_up(bi, 1, 32);
    if (lane == 0) pv = prev;            // carried from previous chunk (t=0 -> -1)

    bool keep = valid && (bi != BLANK) && (bi != pv);
    unsigned mask = (unsigned)__ballot(keep);
    if (keep) {
      int pos = count + __popc(mask & ((1u << lane) - 1u));
      dec[pos] = (float)bi;              // per-wave stores stay ordered vs. the fill
    }
    count += __popc(mask);
    if (valid) suml += m;
    prev = __shfl(bi, 31, 32);
  }

  for (int off = 16; off > 0; off >>= 1) suml += __shfl_xor(suml, off, 32);
  if (lane == 0) out[(size_t)BB * TT + b] = -suml;     // log_prob, shape (B,1)
}

extern "C" void kernel_launch(void* const* d_in, const int* in_sizes, int n_in,
                              void* d_out, int out_size, void* d_ws, size_t ws_size,
                              hipStream_t stream) {
  const float* pred   = (const float*)d_in[0];  // (B,T,V) f32
  const int*   in_len = (const int*)d_in[1];    // (B,1) i32
  float* out = (float*)d_out;                   // [B*T decoded | B log_prob]

  int*   best = (int*)d_ws;                                        // B*T ints
  float* ml   = (float*)((char*)d_ws + sizeof(int) * (size_t)BB * TT);  // B*T floats

  ctc_rowmax_kernel<<<(BB * TT) / ROWS, 256, 0, stream>>>(pred, best, ml);
  ctc_decode_kernel<<<BB, 32, 0, stream>>>(best, ml, in_len, out);
}